// FeatureGen_87084756893712
// MI455X (gfx1250) — compile-verified
//
#include <hip/hip_runtime.h>
#include <hip/hip_bf16.h>
#include <stdint.h>

// ---------------------------------------------------------------------------
// FeatureGen for MI455X (gfx1250, wave32)
//   out[0..2400)       : 5 blocks of (240 means + 240 stds) over 4 segments + full
//   out[2400..3934560) : xi[t,l,c] = 0.5*(x[2t,l,c] + x[2t+1,l,c]), t<16384
// x[t,l,c]: l<10 -> AVG_SET group mean over landmarks; l>=10 -> x_in[t,468+l-10,c]
// Group means computed per-wave with V_WMMA_F32_16X16X4_F32 (M=16 frames,
// K=543 pad 544, N=16 >= 10 groups); frames staged to LDS with CDNA5 async
// global->LDS loads (ASYNCcnt).
// ---------------------------------------------------------------------------

typedef float v2f __attribute__((ext_vector_type(2)));
typedef float v8f __attribute__((ext_vector_type(8)));

#define NIN      543
#define FSTRIDE  1629          // floats per input frame (543*3)
#define FPB      16            // input frames per block
#define BLOCK    256
#define NBLOCKS  2048          // 32768 / FPB
#define STAGE_V4 6516          // FPB*FSTRIDE/4 float4 chunks
#define GM_OFF   (FPB*FSTRIDE + 4)  // pad 4 floats for K=543 dummy read

// Per-lane (n = lane&15) group descriptors: [start, end) and 1/len.
// n >= 10 -> empty range (start 600 > any k) => B column is zero padding.
__constant__ int   c_gs[16]  = {0, 40, 80, 140, 200, 260, 320, 370, 420, 480,
                                600, 600, 600, 600, 600, 600};
__constant__ int   c_ge[16]  = {40, 80, 140, 200, 260, 320, 370, 420, 480, 543,
                                0, 0, 0, 0, 0, 0};
__constant__ float c_gi[16]  = {1.f/40, 1.f/40, 1.f/60, 1.f/60, 1.f/60, 1.f/60,
                                1.f/50, 1.f/50, 1.f/60, 1.f/63,
                                0, 0, 0, 0, 0, 0};

__global__ void fg_zero(float* p, int n) {
  int i = blockIdx.x * blockDim.x + threadIdx.x;
  if (i < n) p[i] = 0.0f;
}

__global__ __launch_bounds__(BLOCK)
void fg_main(const float* __restrict__ x_in,
             float* __restrict__ out_xi,        // -> out + 2400
             float* __restrict__ sums,          // [4][240]
             float* __restrict__ sqs)           // [4][240]
{
  extern __shared__ float smem[];
  float* frames = smem;                 // FPB*FSTRIDE (+4 pad) staged frames
  float* gmeans = smem + GM_OFF;        // [3][16][16] group means

  const int tid = threadIdx.x;
  const int b   = blockIdx.x;

  // ---- Stage 16 contiguous input frames (104256 B) into LDS, async ----
  const float* src = x_in + (size_t)b * (FPB * FSTRIDE);
  const unsigned lds0 = (unsigned)(size_t)(void*)frames;
  for (int i = tid; i < STAGE_V4; i += BLOCK) {
    const float* g = src + i * 4;
    unsigned     l = lds0 + (unsigned)i * 16u;
    asm volatile("global_load_async_to_lds_b128 %0, %1, off"
                 :: "v"(l), "v"(g) : "memory");
  }
  asm volatile("s_wait_asynccnt 0x0" ::: "memory");
  __syncthreads();

  const int wave = tid >> 5;
  const int lane = tid & 31;

  // ---- Phase 1: group means via WMMA f32 16x16x4, one wave per channel ----
  if (wave < 3) {
    const int   ch = wave;
    const int   n  = lane & 15;         // B column / C column (group id)
    const int   hi = lane >> 4;         // lane half selects K pair
    const int   fA = lane & 15;         // A row = frame
    const int   gs = c_gs[n], ge = c_ge[n];
    const float gi = c_gi[n];
    const float* fb = frames + fA * FSTRIDE + ch;

    v8f acc = {};
    for (int k0 = 0; k0 < 544; k0 += 4) {
      const int k = k0 + 2 * hi;        // this lane's K for vgpr0; vgpr1 = k+1
      v2f a, bm;
      a.x  = fb[k * 3];                 // A[frame][k]   (stride 3: channel-interleaved)
      a.y  = fb[k * 3 + 3];             // A[frame][k+1] (k=543 hits LDS pad; B=0 there)
      bm.x = (k     >= gs && k     < ge) ? gi : 0.0f;   // W[k][n]
      bm.y = (k + 1 >= gs && k + 1 < ge) ? gi : 0.0f;   // W[k+1][n]
      acc = __builtin_amdgcn_wmma_f32_16x16x4_f32(
                false, a, false, bm, (short)0, acc, false, false);
    }
    // C layout: VGPR r, lanes0-15 -> M=r, lanes16-31 -> M=r+8; N = lane&15
    #pragma unroll
    for (int r = 0; r < 8; ++r) {
      const int f = r + 8 * hi;
      gmeans[ch * 256 + f * 16 + n] = acc[r];
    }
  }
  __syncthreads();

  // ---- Phase 2: emit resized frames + per-segment partial stats ----
  const int j = tid;                    // feature id = l*3 + c
  if (j < 240) {
    const int l  = j / 3;
    const int ch = j - 3 * l;
    float s_acc = 0.0f, q_acc = 0.0f;
    const int obase = b * 8 * 240 + j;  // output frame base = b*8
    #pragma unroll
    for (int i = 0; i < 8; ++i) {
      const int f0 = 2 * i, f1 = f0 + 1;
      float v0, v1;
      if (l < 10) {
        v0 = gmeans[ch * 256 + f0 * 16 + l];
        v1 = gmeans[ch * 256 + f1 * 16 + l];
      } else {
        const int n = 468 + (l - 10);
        v0 = frames[f0 * FSTRIDE + n * 3 + ch];
        v1 = frames[f1 * FSTRIDE + n * 3 + ch];
      }
      out_xi[obase + i * 240] = 0.5f * (v0 + v1);
      s_acc += v0 + v1;
      q_acc += v0 * v0 + v1 * v1;
    }
    const int seg = b >> 9;             // 512 blocks (8192 frames) per segment
    atomicAdd(&sums[seg * 240 + j], s_acc);
    atomicAdd(&sqs [seg * 240 + j], q_acc);
  }
}

__global__ void fg_finalize(const float* __restrict__ sums,
                            const float* __restrict__ sqs,
                            float* __restrict__ out)
{
  const int j = threadIdx.x;
  if (j >= 240) return;
  float S = 0.0f, Q = 0.0f;
  #pragma unroll
  for (int k = 0; k < 4; ++k) {
    const float s = sums[k * 240 + j];
    const float q = sqs [k * 240 + j];
    float m  = s * (1.0f / 8192.0f);
    float v  = q * (1.0f / 8192.0f) - m * m;
    float sd = sqrtf(fmaxf(v, 0.0f));
    if (!isfinite(m))  m  = 0.0f;
    if (!isfinite(sd)) sd = 0.0f;
    out[k * 480 + j]       = m;
    out[k * 480 + 240 + j] = sd;
    S += s; Q += q;
  }
  float m  = S * (1.0f / 32768.0f);
  float v  = Q * (1.0f / 32768.0f) - m * m;
  float sd = sqrtf(fmaxf(v, 0.0f));
  if (!isfinite(m))  m  = 0.0f;
  if (!isfinite(sd)) sd = 0.0f;
  out[4 * 480 + j]       = m;
  out[4 * 480 + 240 + j] = sd;
}

extern "C" void kernel_launch(void* const* d_in, const int* in_sizes, int n_in,
                              void* d_out, int out_size, void* d_ws, size_t ws_size,
                              hipStream_t stream) {
  (void)in_sizes; (void)n_in; (void)out_size; (void)ws_size;
  const float* x_in = (const float*)d_in[0];
  float* out  = (float*)d_out;
  float* sums = (float*)d_ws;          // 960 floats
  float* sqs  = sums + 960;            // 960 floats

  const size_t shmem = (size_t)(GM_OFF + 768) * sizeof(float); // ~107 KB

  static_assert(GM_OFF + 768 == 16 * 1629 + 4 + 768, "lds layout");

  hipFuncSetAttribute(reinterpret_cast<const void*>(fg_main),
                      hipFuncAttributeMaxDynamicSharedMemorySize, (int)shmem);

  fg_zero<<<(1920 + BLOCK - 1) / BLOCK, BLOCK, 0, stream>>>(sums, 1920);
  fg_main<<<NBLOCKS, BLOCK, shmem, stream>>>(x_in, out + 2400, sums, sqs);
  fg_finalize<<<1, BLOCK, 0, stream>>>(sums, sqs, out);
}